// HypergraphConv_34454227648547
// MI455X (gfx1250) — compile-verified
//
#include <hip/hip_runtime.h>
#include <hip/hip_bf16.h>

typedef __attribute__((ext_vector_type(2))) float v2f;
typedef __attribute__((ext_vector_type(8))) float v8f;

#define IN_F  128
#define OUT_F 128

// ---------------------------------------------------------------------------
// Stage 1: degree histograms.  node_deg counts src[e]; edge_deg counts dst[e]
// (dst values are node ids < N, so an N-sized counter array matches the
// reference's bincount exactly on every index that is ever read back).
// ---------------------------------------------------------------------------
__global__ void hg_degrees(const long long* __restrict__ he,
                           unsigned* __restrict__ ndeg,
                           unsigned* __restrict__ ddeg, int E) {
  int e = blockIdx.x * blockDim.x + threadIdx.x;
  if (e >= E) return;
  atomicAdd(&ndeg[(unsigned)he[e]], 1u);
  atomicAdd(&ddeg[(unsigned)he[(size_t)E + e]], 1u);
}

// ---------------------------------------------------------------------------
// Stage 2: O[N,128] = X[N,128] @ W[128,128] with V_WMMA_F32_16X16X4_F32.
// One wave computes one 16x16 output tile; 8 waves per block cover one row
// tile across all 8 column tiles (W stays resident in L2: it is only 64 KB).
//
// f32 WMMA layouts (wave32):
//   A (16x4, MxK): lane m = lane%16; VGPR0 = K {0 | 2}, VGPR1 = K {1 | 3}
//                  (lanes 0-15 get K=0/1, lanes 16-31 get K=2/3)
//   B (4x16, KxN): lane n = lane%16; VGPR0 = K {0 | 2}, VGPR1 = K {1 | 3}
//   C/D (16x16):   VGPR r -> row r (lanes 0-15) / row r+8 (lanes 16-31)
// ---------------------------------------------------------------------------
__global__ void hg_gemm_wmma(const float* __restrict__ X,
                             const float* __restrict__ W,
                             float* __restrict__ O, int N) {
  const int lane = threadIdx.x & 31;
  const int wave = threadIdx.x >> 5;          // 0..7 -> column tile
  const int ti   = blockIdx.x;                // row tile
  const int m    = lane & 15;
  const int koff = (lane >> 4) << 1;          // 0 for lanes 0-15, 2 for 16-31

  int arow = ti * 16 + m;
  if (arow > N - 1) arow = N - 1;             // clamp reads; stores are guarded
  const float* __restrict__ xrow = X + (size_t)arow * IN_F;
  const int bcol = wave * 16 + m;

  v8f c = {};
#pragma unroll
  for (int k = 0; k < IN_F; k += 4) {
    v2f a, b;
    a.x = xrow[k + koff];
    a.y = xrow[k + koff + 1];
    b.x = W[(size_t)(k + koff) * OUT_F + bcol];
    b.y = W[(size_t)(k + koff + 1) * OUT_F + bcol];
    // (neg_a, A, neg_b, B, c_mod, C, reuse_a, reuse_b)
    c = __builtin_amdgcn_wmma_f32_16x16x4_f32(false, a, false, b,
                                              (short)0, c, false, false);
  }

  const int row0 = ti * 16 + ((lane >> 4) << 3);  // +8 for upper half-wave
#pragma unroll
  for (int r = 0; r < 8; ++r) {
    const int row = row0 + r;
    if (row < N) O[(size_t)row * OUT_F + bcol] = c[r];
  }
}

// ---------------------------------------------------------------------------
// Stage 3: per-edge gather / scale / atomic scatter.  One wave32 per edge,
// each lane owns a float4 slice of the 128-wide row (32 lanes x 4 = 128).
// Gathers hit L2 (mm is 25.6 MB < 192 MB L2); scatter uses f32 L2 atomics.
// ---------------------------------------------------------------------------
__global__ void hg_edge_scatter(const long long* __restrict__ he,
                                const float* __restrict__ ew,
                                const unsigned* __restrict__ ndeg,
                                const unsigned* __restrict__ ddeg,
                                const float* __restrict__ mm,
                                float* __restrict__ out, int E) {
  const int gw   = (int)(((size_t)blockIdx.x * blockDim.x + threadIdx.x) >> 5);
  const int lane = threadIdx.x & 31;
  if (gw >= E) return;

  const int s = (int)he[gw];                  // segment (output row)
  const int d = (int)he[(size_t)E + gw];      // gather row
  const float nd  = (float)ndeg[s];
  const float ed  = (float)ddeg[d];
  const float val = rsqrtf(nd) * (1.0f / (ed + 1e-5f)) * ew[gw];

  const float4 g = ((const float4*)(mm + (size_t)d * OUT_F))[lane];
  float* dst = out + (size_t)s * OUT_F + lane * 4;
  atomicAdd(dst + 0, g.x * val);
  atomicAdd(dst + 1, g.y * val);
  atomicAdd(dst + 2, g.z * val);
  atomicAdd(dst + 3, g.w * val);
}

// ---------------------------------------------------------------------------
// Inputs (setup_inputs order): x[N*128] f32, hyperedge_index[2*E] i64,
// hyperedge_weight[E] f32, weight[128*128] f32.  Output: [N*128] f32.
// Workspace: ndeg[N] u32 | ddeg[N] u32 | (256B-aligned) mm[N*128] f32.
// ---------------------------------------------------------------------------
extern "C" void kernel_launch(void* const* d_in, const int* in_sizes, int n_in,
                              void* d_out, int out_size, void* d_ws, size_t ws_size,
                              hipStream_t stream) {
  const float*     x  = (const float*)d_in[0];
  const long long* he = (const long long*)d_in[1];
  const float*     ew = (const float*)d_in[2];
  const float*     w  = (const float*)d_in[3];
  float* out = (float*)d_out;

  const int N = in_sizes[0] / IN_F;
  const int E = in_sizes[2];

  unsigned* ndeg = (unsigned*)d_ws;
  unsigned* ddeg = ndeg + N;
  const size_t mm_off = (((size_t)2 * N * sizeof(unsigned)) + 255) & ~(size_t)255;
  float* mm = (float*)((char*)d_ws + mm_off);

  // Zero the accumulation buffers (d_out / d_ws are poisoned by the harness).
  hipMemsetAsync(d_out, 0, (size_t)out_size * sizeof(float), stream);
  hipMemsetAsync(d_ws, 0, (size_t)2 * N * sizeof(unsigned), stream);

  hg_degrees<<<(E + 255) / 256, 256, 0, stream>>>(he, ndeg, ddeg, E);
  hg_gemm_wmma<<<(N + 15) / 16, 256, 0, stream>>>(x, w, mm, N);
  hg_edge_scatter<<<(E + 7) / 8, 256, 0, stream>>>(he, ew, ndeg, ddeg, mm, out, E);
}